// CDBNE_14302241095766
// MI455X (gfx1250) — compile-verified
//
#include <hip/hip_runtime.h>
#include <math.h>

typedef float v2f __attribute__((ext_vector_type(2)));
typedef float v8f __attribute__((ext_vector_type(8)));

#define WAVE 32

// ---------- order-preserving float<->uint encode for atomic max ----------
__device__ __forceinline__ unsigned enc_f(float f) {
    unsigned u = __float_as_uint(f);
    return (u & 0x80000000u) ? ~u : (u | 0x80000000u);
}
__device__ __forceinline__ float dec_f(unsigned u) {
    return (u & 0x80000000u) ? __uint_as_float(u & 0x7fffffffu)
                             : __uint_as_float(~u);
}
// enc(-inf) = ~0xFF800000 = 0x007FFFFF
#define ENC_NEG_INF 0x007FFFFFu

// ---------- L2 row normalize (wave per row) ----------
__global__ void k_l2norm(const float* __restrict__ in, float* __restrict__ out,
                         int n, int d) {
    int wid  = blockIdx.x * (blockDim.x / WAVE) + (threadIdx.x / WAVE);
    int lane = threadIdx.x & (WAVE - 1);
    if (wid >= n) return;
    const float* row = in + (size_t)wid * d;
    float s = 0.f;
    for (int j = lane; j < d; j += WAVE) { float v = row[j]; s += v * v; }
    for (int off = 16; off; off >>= 1) s += __shfl_down(s, off, WAVE);
    s = __shfl(s, 0, WAVE);
    float inv = 1.0f / fmaxf(sqrtf(s), 1e-12f);
    float* orow = out + (size_t)wid * d;
    for (int j = lane; j < d; j += WAVE) orow[j] = row[j] * inv;
}

// ---------- fp32 WMMA GEMM: H[n x DOUT] = A[n x DIN] * W[DIN x DOUT] ----------
// 256-thread block = 8 waves; block covers 128 rows of A (16 per wave).
// W staged through LDS in 32-row K-chunks, stored PAIR-SWIZZLED:
//   sW[kpair][col][0..1] = W[2*kpair][col], W[2*kpair+1][col]
// so a B fragment (values K=ka and K=ka+1 for one column) is a single
// aligned ds_load_b64 directly into the WMMA operand VGPR pair (no moves).
// Row stride LDW = 2*DOUT+32 floats puts half 0 (lanes 0-15) and half 1
// (lanes 16-31) on disjoint 32-bank sets -> conflict-free b64 reads.
// Each wave holds all DOUT/16 accumulator tiles in registers; the A
// fragment is one aligned b64 global load, mask-multiplied (no exec
// divergence, WMMA sees EXEC all-ones).
template <int DIN, int DOUT>
__global__ __launch_bounds__(256)
void k_gemm_wmma(const float* __restrict__ A, const float* __restrict__ W,
                 float* __restrict__ H, int n) {
    constexpr int KC  = 32;              // K-chunk staged in LDS
    constexpr int CT  = DOUT / 16;       // column tiles per wave
    constexpr int LDW = 2 * DOUT + 32;   // padded floats per k-pair row
    __shared__ float sW[(KC / 2) * LDW]; // <= 18 KB

    const int tid  = threadIdx.x;
    const int wave = tid / WAVE;
    const int lane = tid % WAVE;
    const int half = lane >> 4;          // 0|1 : K sub-pair
    const int ml   = lane & 15;

    const int rowBase = blockIdx.x * 128 + wave * 16;
    const int arow    = rowBase + ml;
    const float amask = (arow < n) ? 1.f : 0.f;
    const float* Arow = A + (size_t)(arow < n ? arow : 0) * DIN;

    v8f acc[CT];
#pragma unroll
    for (int ct = 0; ct < CT; ++ct) acc[ct] = (v8f){};

    for (int k0 = 0; k0 < DIN; k0 += KC) {
        // cooperative pair-swizzled copy: W[k0:k0+KC, :] -> LDS
        const float* Wg = W + (size_t)k0 * DOUT;
#pragma unroll
        for (int i = tid; i < (KC / 2) * DOUT; i += 256) {
            int kp  = i / DOUT;          // power-of-2 division
            int col = i % DOUT;
            v2f w;
            w[0] = Wg[(2 * kp)     * DOUT + col];
            w[1] = Wg[(2 * kp + 1) * DOUT + col];
            *(v2f*)&sW[kp * LDW + col * 2] = w;
        }
        __syncthreads();

#pragma unroll
        for (int kk = 0; kk < KC; kk += 4) {
            v2f a = *(const v2f*)(Arow + k0 + kk + 2 * half);
            a[0] *= amask;
            a[1] *= amask;
            const int brow = (kk / 2 + half) * LDW;
#pragma unroll
            for (int ct = 0; ct < CT; ++ct) {
                v2f b = *(const v2f*)&sW[brow + (ct * 16 + ml) * 2];
                acc[ct] = __builtin_amdgcn_wmma_f32_16x16x4_f32(
                    false, a, false, b, (short)0, acc[ct], false, false);
            }
        }
        __syncthreads();
    }

#pragma unroll
    for (int ct = 0; ct < CT; ++ct) {
#pragma unroll
        for (int r = 0; r < 8; ++r) {
            int orow = rowBase + r + 8 * half;
            if (orow < n) H[(size_t)orow * DOUT + ct * 16 + ml] = acc[ct][r];
        }
    }
}

// ---------- per-node attention coefficients: as[i]=h_i.a_src, ad[i]=h_i.a_dst ----------
__global__ void k_attn(const float* __restrict__ H, const float* __restrict__ asrc,
                       const float* __restrict__ adst, float* __restrict__ als,
                       float* __restrict__ ald, int n, int d) {
    int wid  = blockIdx.x * (blockDim.x / WAVE) + (threadIdx.x / WAVE);
    int lane = threadIdx.x & (WAVE - 1);
    if (wid >= n) return;
    const float* row = H + (size_t)wid * d;
    float s0 = 0.f, s1 = 0.f;
    for (int j = lane; j < d; j += WAVE) {
        float v = row[j];
        s0 += v * asrc[j];
        s1 += v * adst[j];
    }
    for (int off = 16; off; off >>= 1) {
        s0 += __shfl_down(s0, off, WAVE);
        s1 += __shfl_down(s1, off, WAVE);
    }
    if (lane == 0) { als[wid] = s0; ald[wid] = s1; }
}

// ---------- init segment buffers ----------
__global__ void k_init(unsigned* __restrict__ m, float* __restrict__ den,
                       float* __restrict__ out, int n, int tot) {
    int i = blockIdx.x * blockDim.x + threadIdx.x;
    if (i < n) { m[i] = ENC_NEG_INF; den[i] = 0.f; }
    if (i < tot) out[i] = 0.f;
}

// ---------- edge pass 1: leaky-relu logits + segment max over dst ----------
__global__ void k_edge_logits(const int* __restrict__ ei, const float* __restrict__ als,
                              const float* __restrict__ ald, float* __restrict__ elog,
                              unsigned* __restrict__ m, int E, int n) {
    int e = blockIdx.x * blockDim.x + threadIdx.x;
    int Etot = E + n;
    if (e >= Etot) return;
    int s = (e < E) ? ei[e]     : (e - E);
    int t = (e < E) ? ei[E + e] : (e - E);
    float l = als[s] + ald[t];
    l = (l > 0.f) ? l : 0.2f * l;
    elog[e] = l;
    atomicMax(&m[t], enc_f(l));
}

// ---------- edge pass 2: exp(l - max) and segment sum ----------
__global__ void k_edge_exp(const int* __restrict__ ei, float* __restrict__ elog,
                           const unsigned* __restrict__ m, float* __restrict__ den,
                           int E, int n) {
    int e = blockIdx.x * blockDim.x + threadIdx.x;
    int Etot = E + n;
    if (e >= Etot) return;
    int t = (e < E) ? ei[E + e] : (e - E);
    float ex = __expf(elog[e] - dec_f(m[t]));
    elog[e] = ex;
    atomicAdd(&den[t], ex);
}

// ---------- edge pass 3: out[dst] += h[src] * alpha  (wave per edge) ----------
__global__ void k_edge_aggr(const int* __restrict__ ei, const float* __restrict__ H,
                            const float* __restrict__ elog, const float* __restrict__ den,
                            float* __restrict__ out, int E, int n, int d) {
    long long gid = (long long)blockIdx.x * blockDim.x + threadIdx.x;
    int e    = (int)(gid / WAVE);
    int lane = (int)(gid % WAVE);
    int Etot = E + n;
    if (e >= Etot) return;
    int s = (e < E) ? ei[e]     : (e - E);
    int t = (e < E) ? ei[E + e] : (e - E);
    float alpha = elog[e] / den[t];
    const float* hrow = H + (size_t)s * d;
    float* orow = out + (size_t)t * d;
    for (int j = lane; j < d; j += WAVE)
        atomicAdd(&orow[j], hrow[j] * alpha);
}

// ---------- epilogue: out = act(in + bias) ----------
__global__ void k_bias_act(const float* __restrict__ in, const float* __restrict__ b,
                           float* __restrict__ out, int n, int d, int relu) {
    long long i = (long long)blockIdx.x * blockDim.x + threadIdx.x;
    if (i >= (long long)n * d) return;
    int j = (int)(i % d);
    float v = in[i] + b[j];
    if (relu) v = fmaxf(v, 0.f);
    out[i] = v;
}

// ---------- Student-t cluster assignment (ALPHA = 1) ----------
__global__ void k_cluster(const float* __restrict__ z, const float* __restrict__ cl,
                          float* __restrict__ q, int n, int d, int K) {
    int i = blockIdx.x * blockDim.x + threadIdx.x;
    if (i >= n) return;
    const float* zr = z + (size_t)i * d;
    float z2 = 0.f;
    for (int j = 0; j < d; ++j) z2 += zr[j] * zr[j];
    float qs[16];
    float ssum = 0.f;
    for (int k = 0; k < K; ++k) {
        const float* cr = cl + (size_t)k * d;
        float c2 = 0.f, dot = 0.f;
        for (int j = 0; j < d; ++j) { float cv = cr[j]; c2 += cv * cv; dot += zr[j] * cv; }
        float dist = z2 + c2 - 2.0f * dot;
        float qq = 1.0f / (1.0f + dist) + 1e-7f;   // exponent (ALPHA+1)/2 == 1
        qs[k] = qq;
        ssum += qq;
    }
    float inv = 1.0f / ssum;
    for (int k = 0; k < K; ++k) q[(size_t)i * K + k] = qs[k] * inv;
}

extern "C" void kernel_launch(void* const* d_in, const int* in_sizes, int n_in,
                              void* d_out, int out_size, void* d_ws, size_t ws_size,
                              hipStream_t stream) {
    const int IN_D = 128, HID_D = 128, EMB_D = 64, K = 16;
    const int N = in_sizes[0] / IN_D;
    const int E = in_sizes[1] / 2;
    const int Etot = E + N;

    const float* x  = (const float*)d_in[0];
    const int*   ei = (const int*)d_in[1];
    const float* W1 = (const float*)d_in[2];
    const float* a1s = (const float*)d_in[3];
    const float* a1d = (const float*)d_in[4];
    const float* b1  = (const float*)d_in[5];
    const float* W2 = (const float*)d_in[6];
    const float* a2s = (const float*)d_in[7];
    const float* a2d = (const float*)d_in[8];
    const float* b2  = (const float*)d_in[9];
    const float* W3 = (const float*)d_in[10];
    const float* a3s = (const float*)d_in[11];
    const float* a3d = (const float*)d_in[12];
    const float* b3  = (const float*)d_in[13];
    const float* W4 = (const float*)d_in[14];
    const float* a4s = (const float*)d_in[15];
    const float* a4d = (const float*)d_in[16];
    const float* b4  = (const float*)d_in[17];
    const float* cluster = (const float*)d_in[18];

    float* out_z    = (float*)d_out;                 // [N, 64]
    float* out_xhat = out_z + (size_t)N * EMB_D;     // [N, 128]
    float* out_q    = out_xhat + (size_t)N * IN_D;   // [N, 16]

    // workspace carve-up (floats)
    float* wsf  = (float*)d_ws;
    float* buf0 = wsf;                               // N*128 : layer input
    float* bufH = buf0 + (size_t)N * 128;            // N*128 : h = x W
    float* bufO = bufH + (size_t)N * 128;            // N*128 : aggregated out
    float*    v_as  = bufO + (size_t)N * 128;        // N
    float*    v_ad  = v_as + N;                      // N
    unsigned* v_m   = (unsigned*)(v_ad + N);         // N (encoded max)
    float*    v_den = (float*)(v_m + N);             // N
    float*    elog  = v_den + N;                     // E+N (logits -> exp)

    const int TB = 256;
    const int WPB = TB / WAVE;

    auto launch_gemm = [&](const float* X, const float* W, int din, int dout) {
        dim3 g((N + 127) / 128);
        if (din == 128 && dout == 128)
            k_gemm_wmma<128, 128><<<g, 256, 0, stream>>>(X, W, bufH, N);
        else if (din == 128 && dout == 64)
            k_gemm_wmma<128, 64><<<g, 256, 0, stream>>>(X, W, bufH, N);
        else
            k_gemm_wmma<64, 128><<<g, 256, 0, stream>>>(X, W, bufH, N);
    };

    auto gat = [&](const float* X, const float* W, const float* a_s,
                   const float* a_d, int din, int dout) {
        launch_gemm(X, W, din, dout);
        k_attn<<<(N + WPB - 1) / WPB, TB, 0, stream>>>(bufH, a_s, a_d, v_as, v_ad, N, dout);
        int tot = N * dout;
        k_init<<<(tot + TB - 1) / TB, TB, 0, stream>>>(v_m, v_den, bufO, N, tot);
        k_edge_logits<<<(Etot + TB - 1) / TB, TB, 0, stream>>>(ei, v_as, v_ad, elog, v_m, E, N);
        k_edge_exp<<<(Etot + TB - 1) / TB, TB, 0, stream>>>(ei, elog, v_m, v_den, E, N);
        long long th = (long long)Etot * WAVE;
        k_edge_aggr<<<(unsigned)((th + TB - 1) / TB), TB, 0, stream>>>(
            ei, bufH, elog, v_den, bufO, E, N, dout);
    };

    auto epi = [&](const float* in, const float* b, float* out, int d, int relu) {
        long long tot = (long long)N * d;
        k_bias_act<<<(unsigned)((tot + TB - 1) / TB), TB, 0, stream>>>(in, b, out, N, d, relu);
    };

    // xn = l2norm(x)
    k_l2norm<<<(N + WPB - 1) / WPB, TB, 0, stream>>>(x, buf0, N, IN_D);

    // layer 1: h1 = relu(GAT(xn, W1)) -> buf0
    gat(buf0, W1, a1s, a1d, IN_D, HID_D);
    epi(bufO, b1, buf0, HID_D, 1);

    // layer 2: z = l2norm(GAT(h1, W2)) -> out_z
    gat(buf0, W2, a2s, a2d, HID_D, EMB_D);
    epi(bufO, b2, bufH, EMB_D, 0);           // z_pre (bufH free after aggr)
    k_l2norm<<<(N + WPB - 1) / WPB, TB, 0, stream>>>(bufH, out_z, N, EMB_D);

    // layer 3: d1 = relu(GAT(z, W3)) -> buf0
    gat(out_z, W3, a3s, a3d, EMB_D, HID_D);
    epi(bufO, b3, buf0, HID_D, 1);

    // layer 4: x_hat = GAT(d1, W4) -> out_xhat
    gat(buf0, W4, a4s, a4d, HID_D, IN_D);
    epi(bufO, b4, out_xhat, IN_D, 0);

    // q from z
    k_cluster<<<(N + TB - 1) / TB, TB, 0, stream>>>(out_z, cluster, out_q, N, EMB_D, K);
}